// InterpositionAttention_16286515986685
// MI455X (gfx1250) — compile-verified
//
#include <hip/hip_runtime.h>
#include <hip/hip_bf16.h>
#include <math.h>

// Problem constants (from reference setup_inputs)
static constexpr int B  = 16;
static constexpr int C  = 64;
static constexpr int N  = 2000;   // 125 tiles of 16
static constexpr int OC = 64;

typedef float v2f __attribute__((ext_vector_type(2)));
typedef float v8f __attribute__((ext_vector_type(8)));

__device__ __forceinline__ float lrelu(float x) {
    return x > 0.0f ? x : 0.01f * x;
}

// ---------------------------------------------------------------------------
// Kernel 1: h[b][n][oc] = sum_c inp[b][c][n] * W[c][oc]
//           f1[b*N+n]   = sum_oc h*a1[oc], f2 likewise (a2 = a[OC+oc])
// One 64-thread block per (b,n); x-column staged in LDS, W streamed from L2.
// ---------------------------------------------------------------------------
__global__ __launch_bounds__(64) void k_proj(const float* __restrict__ inp,
                                             const float* __restrict__ W,
                                             const float* __restrict__ a,
                                             float* __restrict__ h,
                                             float* __restrict__ f1,
                                             float* __restrict__ f2) {
    __shared__ float xs[C];
    __shared__ float red[OC];
    const int bn = blockIdx.x;          // = b*N + n
    const int b  = bn / N;
    const int n  = bn % N;
    const int oc = threadIdx.x;         // 0..63 (also used as c index for staging)

    xs[oc] = inp[(size_t)b * C * N + (size_t)oc * N + n];
    __syncthreads();

    float acc = 0.0f;
#pragma unroll
    for (int c = 0; c < C; ++c)
        acc = fmaf(xs[c], W[c * OC + oc], acc);
    h[(size_t)bn * OC + oc] = acc;

    // f1 reduction
    red[oc] = acc * a[oc];
    __syncthreads();
#pragma unroll
    for (int s = 32; s > 0; s >>= 1) {
        if (oc < s) red[oc] += red[oc + s];
        __syncthreads();
    }
    if (oc == 0) f1[bn] = red[0];
    __syncthreads();

    // f2 reduction
    red[oc] = acc * a[OC + oc];
    __syncthreads();
#pragma unroll
    for (int s = 32; s > 0; s >>= 1) {
        if (oc < s) red[oc] += red[oc + s];
        __syncthreads();
    }
    if (oc == 0) f2[bn] = red[0];
}

// ---------------------------------------------------------------------------
// Kernel 2: per-batch max of f2 (lrelu is monotone => row max = lrelu(f1_i+max f2))
// ---------------------------------------------------------------------------
__global__ __launch_bounds__(256) void k_f2max(const float* __restrict__ f2,
                                               float* __restrict__ m2) {
    __shared__ float s[256];
    const int b = blockIdx.x;
    const int t = threadIdx.x;
    float mx = -INFINITY;
    for (int j = t; j < N; j += 256) mx = fmaxf(mx, f2[b * N + j]);
    s[t] = mx;
    __syncthreads();
#pragma unroll
    for (int w = 128; w > 0; w >>= 1) {
        if (t < w) s[t] = fmaxf(s[t], s[t + w]);
        __syncthreads();
    }
    if (t == 0) m2[b] = s[0];
}

// ---------------------------------------------------------------------------
// Kernel 3: per-row softmax stats. m_i = lrelu(f1_i + m2_b);
//           invl_i = 1 / sum_j exp(lrelu(f1_i + f2_j) - m_i)
// One wave32 per row, 8 waves per block.
// ---------------------------------------------------------------------------
__global__ __launch_bounds__(256) void k_rowstats(const float* __restrict__ f1,
                                                  const float* __restrict__ f2,
                                                  const float* __restrict__ m2,
                                                  float* __restrict__ mrow,
                                                  float* __restrict__ invl) {
    const int wave = threadIdx.x >> 5;
    const int lane = threadIdx.x & 31;
    const int row  = blockIdx.x * 8 + wave;   // 0 .. B*N-1
    const int b    = row / N;

    const float f1i = f1[row];
    const float mi  = lrelu(f1i + m2[b]);
    float s = 0.0f;
    for (int j = lane; j < N; j += 32)
        s += __expf(lrelu(f1i + f2[b * N + j]) - mi);
#pragma unroll
    for (int off = 16; off > 0; off >>= 1)
        s += __shfl_down(s, off, 32);
    if (lane == 0) {
        mrow[row] = mi;
        invl[row] = 1.0f / s;
    }
}

// ---------------------------------------------------------------------------
// Kernel 4: flash-style attention with V_WMMA_F32_16X16X4_F32.
// One wave32 per (batch, 16-row tile). For each 16-col J tile:
//   P[m][j] = exp(lrelu(f1_m + f2_j) - m_row[m])   (unnormalized)
//   acc[t] += P(16x16) @ H(16x16)  via 4 K-chunks x 4 OC-tiles = 16 WMMAs
// A-fragment (16x4 f32): lane L -> row M=L&15; vgpr0 holds K=2*(L>>4),
// vgpr1 holds K=2*(L>>4)+1 (ISA 7.12.2). B-fragment (4x16) mirrors that
// with col N=L&15. C/D: vgpr v -> row v+8*(L>>4), col L&15.
// ---------------------------------------------------------------------------
__global__ __launch_bounds__(32) void k_attn(const float* __restrict__ h,
                                             const float* __restrict__ f1,
                                             const float* __restrict__ f2,
                                             const float* __restrict__ mrow,
                                             const float* __restrict__ invl,
                                             float* __restrict__ out) {
    const int lane = threadIdx.x;
    const int half = lane >> 4;     // 0 or 1
    const int l16  = lane & 15;
    const int b    = blockIdx.y;
    const int I    = blockIdx.x * 16;

    const float* __restrict__ hb  = h  + (size_t)b * N * OC;
    const float* __restrict__ f2b = f2 + b * N;
    const float  f1r = f1[b * N + I + l16];     // row for this lane's A-fragment
    const float  mr  = mrow[b * N + I + l16];

    v8f acc[4];
#pragma unroll
    for (int t = 0; t < 4; ++t) acc[t] = (v8f){};

    for (int J = 0; J < N; J += 16) {
        const float* __restrict__ ht = hb + (size_t)J * OC;
        if (J + 16 < N)  // warm L1/WGP$ for the next H tile (global_prefetch_b8)
            __builtin_prefetch((const void*)(hb + (size_t)(J + 16) * OC + lane * 4), 0, 1);

#pragma unroll
        for (int kk = 0; kk < 4; ++kk) {
            const int k0 = kk * 4 + 2 * half;   // this lane's K indices: k0, k0+1
            // Build P fragment directly in A-layout registers.
            float e0 = lrelu(f1r + f2b[J + k0]);
            float e1 = lrelu(f1r + f2b[J + k0 + 1]);
            v2f afrag;
            afrag.x = __expf(e0 - mr);
            afrag.y = __expf(e1 - mr);

            const float* __restrict__ hr = ht + (size_t)k0 * OC;
#pragma unroll
            for (int t = 0; t < 4; ++t) {
                v2f bfrag;
                bfrag.x = hr[t * 16 + l16];        // row J+k0
                bfrag.y = hr[OC + t * 16 + l16];   // row J+k0+1
                acc[t] = __builtin_amdgcn_wmma_f32_16x16x4_f32(
                    /*neg_a=*/false, afrag, /*neg_b=*/false, bfrag,
                    /*c_mod=*/(short)0, acc[t],
                    /*reuse_a=*/false, /*reuse_b=*/false);
            }
        }
    }

    // Normalize by 1/l, relu, write transposed output out[b][oc][n].
    const float* __restrict__ ilp = invl + b * N + I + 8 * half;
    float il[8];
#pragma unroll
    for (int v = 0; v < 8; ++v) il[v] = ilp[v];

    float* __restrict__ ob = out + (size_t)b * OC * N;
#pragma unroll
    for (int t = 0; t < 4; ++t) {
#pragma unroll
        for (int v = 0; v < 8; ++v) {
            float val = acc[t][v] * il[v];
            val = val > 0.0f ? val : 0.0f;
            ob[(size_t)(t * 16 + l16) * N + I + 8 * half + v] = val;
        }
    }
}

// ---------------------------------------------------------------------------
extern "C" void kernel_launch(void* const* d_in, const int* in_sizes, int n_in,
                              void* d_out, int out_size, void* d_ws, size_t ws_size,
                              hipStream_t stream) {
    const float* inp = (const float*)d_in[0];   // (B, C, N)
    const float* W   = (const float*)d_in[1];   // (C, OC)
    const float* a   = (const float*)d_in[2];   // (2*OC, 1)
    // d_in[3] = GL is provably unused: softmax output is strictly positive,
    // so the adj>0 mask in the reference selects `e` everywhere.
    float* out = (float*)d_out;                 // (B, OC, N)

    float* ws   = (float*)d_ws;
    float* h    = ws;                           // B*N*OC  = 2,048,000 f32
    float* f1   = h    + (size_t)B * N * OC;    // B*N     =    32,000
    float* f2   = f1   + (size_t)B * N;         // B*N
    float* m2   = f2   + (size_t)B * N;         // B (padded to 32)
    float* mrow = m2   + 32;                    // B*N
    float* invl = mrow + (size_t)B * N;         // B*N     -> ~8.7 MB total

    k_proj    <<<dim3(B * N),       64, 0, stream>>>(inp, W, a, h, f1, f2);
    k_f2max   <<<dim3(B),          256, 0, stream>>>(f2, m2);
    k_rowstats<<<dim3((B * N) / 8), 256, 0, stream>>>(f1, f2, m2, mrow, invl);
    k_attn    <<<dim3(N / 16, B),   32, 0, stream>>>(h, f1, f2, mrow, invl, out);
}